// w2w_50551765074045
// MI455X (gfx1250) — compile-verified
//
#include <hip/hip_runtime.h>
#include <hip/hip_bf16.h>
#include <math.h>

// word2vec negative-sampling BCE loss for MI455X (gfx1250, wave32).
// Memory-bound (~1.8 GB gathered reads, 23.3 TB/s -> ~78us floor).
// Dot products via V_WMMA_F32_16X16X4_F32 diagonal trick:
// per 16-row batch tile, C = X_tile * Y_stack^T ; diag(C) = logits.
// Row indices are software-pipelined 2 deep; the next Y row is pulled in
// with near-scope global_prefetch_b8 (counter-free) to cover latency.

#define DIM   128
#define KTOT  26          // 1 positive + 25 negatives
#define NCHUNK (DIM / 4)  // 32 K=4 WMMA steps per 128-dot
#define WAVES_PER_BLOCK 8

typedef float v2f __attribute__((ext_vector_type(2)));
typedef float v8f __attribute__((ext_vector_type(8)));

__global__ void w2v_zero_out(float* out) {
  if (threadIdx.x == 0 && blockIdx.x == 0) out[0] = 0.0f;
}

__device__ __forceinline__ float bce_term(float logit, float t) {
  // BCEWithLogits, stable: max(l,0) - l*t + log1p(exp(-|l|))
  return fmaxf(logit, 0.0f) - logit * t + log1pf(expf(-fabsf(logit)));
}

__global__ __launch_bounds__(WAVES_PER_BLOCK * 32)
void w2v_loss_wmma(const float* __restrict__ in_emb,   // [VOCAB,128]
                   const float* __restrict__ out_emb,  // [VOCAB,128]
                   const int*   __restrict__ in_ids,   // [B]
                   const int*   __restrict__ out_ids,  // [B, KTOT]
                   float* __restrict__ out,            // [1]
                   int batch) {
  __shared__ float s_acc;
  if (threadIdx.x == 0) s_acc = 0.0f;
  __syncthreads();

  const int lane = threadIdx.x & 31;
  const int wave = threadIdx.x >> 5;
  const int tile = blockIdx.x * WAVES_PER_BLOCK + wave;   // wave-uniform
  const int m    = lane & 15;      // which of the 16 batch rows this lane feeds
  const int hi   = lane >> 4;      // half-wave: K-offset 0 or 2 within chunk

  float lacc = 0.0f;

  if (tile * 16 < batch) {         // uniform branch: EXEC stays all-ones
    const int b    = tile * 16 + m;
    const int base = b * KTOT;

    const int xrow = in_ids[b] * DIM;

    // ---- A operand: X tile, 16x128 f32, register-resident (64 VGPRs).
    // A layout (16x4 per chunk): lane m holds A[m, 2*hi + {0,1}] in vgpr {0,1}
    // -> one b64 load per lane per chunk at row offset 4c + 2*hi.
    v2f a[NCHUNK];
#pragma unroll
    for (int c = 0; c < NCHUNK; ++c) {
      a[c] = *(const v2f*)(in_emb + xrow + 4 * c + 2 * hi);
    }

    // Software-pipeline row indices 2 deep: i0 = row k, i1 = row k+1.
    int i0 = out_ids[base];
    int i1 = out_ids[base + ((KTOT > 1) ? 1 : 0)];

    // Warm the near cache for row 0 (64 cachelines of the 16-row set, 2 ops:
    // lane (m,hi) covers its own row at byte offsets hi*256 and hi*256+128).
    {
      const float* r0 = out_emb + i0 * DIM + hi * 64;
      __builtin_prefetch(r0, 0, 3);        // lines 0/2 of each row, WGP-near
      __builtin_prefetch(r0 + 32, 0, 3);   // lines 1/3 of each row
    }

    for (int k = 0; k < KTOT; ++k) {
      // Issue index load for row k+2 first: oldest load of this iteration,
      // so next trip's wait on it is cheap (no loadcnt drain).
      const int kn = (k + 2 < KTOT) ? (k + 2) : (KTOT - 1);
      const int i2 = out_ids[base + kn];

      // Counter-free near-scope prefetch of row k+1 (index i1 resident).
      const float* rn = out_emb + i1 * DIM + hi * 64;
      __builtin_prefetch(rn, 0, 3);
      __builtin_prefetch(rn + 32, 0, 3);

      const float* yrow = out_emb + i0 * DIM + 2 * hi;
      v8f acc = {};   // C := 0, accumulate 32 chained K=4 WMMAs
#pragma unroll
      for (int c = 0; c < NCHUNK; ++c) {
        // B operand (4x16): lane m holds B[2*hi + {0,1}, m] = Y[b_m,k, 4c+2hi+{0,1}]
        v2f bm = *(const v2f*)(yrow + 4 * c);
        acc = __builtin_amdgcn_wmma_f32_16x16x4_f32(
            /*neg_a=*/false, a[c], /*neg_b=*/false, bm,
            /*c_mod=*/(short)0, acc, /*reuse_a=*/false, /*reuse_b=*/false);
      }

      // diag(C): C vgpr r -> lanes 0..15: (M=r,N=lane); lanes 16..31: (M=r+8,N=lane-16)
      // (i,i) lives at lane r (i=r<8) and lane r+24 (i=r+8). Branch-free select chain.
      float diag = 0.0f;
#pragma unroll
      for (int r = 0; r < 8; ++r) {
        diag = ((lane == r) || (lane == r + 24)) ? acc[r] : diag;
      }
      const bool active = (lane < 8) || (lane >= 24);  // 16 diag-holding lanes
      const float t = (k == 0) ? 1.0f : -1.0f;
      lacc += active ? bce_term(diag, t) : 0.0f;  // select, no divergence near WMMA

      i0 = i1;
      i1 = i2;
    }
  }

  // wave32 butterfly reduction
#pragma unroll
  for (int off = 16; off > 0; off >>= 1) {
    lacc += __shfl_xor(lacc, off, 32);
  }
  if (lane == 0) atomicAdd(&s_acc, lacc);   // ds_add_f32, one per wave
  __syncthreads();
  if (threadIdx.x == 0) atomicAdd(out, s_acc);  // one global atomic per block
}

extern "C" void kernel_launch(void* const* d_in, const int* in_sizes, int n_in,
                              void* d_out, int out_size, void* d_ws, size_t ws_size,
                              hipStream_t stream) {
  const float* in_emb  = (const float*)d_in[0];
  const float* out_emb = (const float*)d_in[1];
  const int*   in_ids  = (const int*)d_in[2];
  const int*   out_ids = (const int*)d_in[3];
  float* out = (float*)d_out;

  const int batch = in_sizes[2];                       // 131072
  const int tiles = (batch + 15) / 16;                 // 8192 wave-tiles
  const int blocks = (tiles + WAVES_PER_BLOCK - 1) / WAVES_PER_BLOCK;  // 1024

  w2v_zero_out<<<1, 1, 0, stream>>>(out);
  w2v_loss_wmma<<<blocks, WAVES_PER_BLOCK * 32, 0, stream>>>(
      in_emb, out_emb, in_ids, out_ids, out, batch);
}